// MulEdgeSoftmax_20968030339289
// MI455X (gfx1250) — compile-verified
//
#include <hip/hip_runtime.h>
#include <hip/hip_bf16.h>
#include <math.h>

// MulEdgeSoftmax: per-head rescale -> exp -> segment-sum by row -> normalize.
// Memory-bound: ~920 MB total traffic -> ~40us at 23.3 TB/s HBM on MI455X.

typedef float v4f __attribute__((ext_vector_type(4)));

#define N_NODES 100000
#define N_HEADS 8
#define STREAM_BLOCKS 4096
#define BLOCK 256

// Order-preserving float -> u32 encoding so we can use native u32 atomicMax
// (avoids a CAS loop for float max). key(a) < key(b) <=> a < b.
__device__ __forceinline__ unsigned enc_f32(float f) {
    unsigned u = __float_as_uint(f);
    return (u & 0x80000000u) ? ~u : (u | 0x80000000u);
}
__device__ __forceinline__ float dec_f32(unsigned k) {
    return (k & 0x80000000u) ? __uint_as_float(k ^ 0x80000000u)
                             : __uint_as_float(~k);
}

// ---------------------------------------------------------------- init ------
__global__ void k_init(float* __restrict__ node_sum, unsigned* __restrict__ keys, int n) {
    int i = blockIdx.x * blockDim.x + threadIdx.x;
    if (i < n) node_sum[i] = 0.0f;
    if (i < N_HEADS) keys[i] = 0u;  // decodes to -NaN; any real float beats it
}

// ------------------------------------------------------- per-head max -------
__global__ void k_max(const float* __restrict__ ev, unsigned* __restrict__ keys, int E) {
    const long long tid = blockIdx.x * blockDim.x + threadIdx.x;
    const long long stride = (long long)gridDim.x * blockDim.x;
    const v4f* p = (const v4f*)ev;

    v4f m0 = (v4f)(-INFINITY);
    v4f m1 = (v4f)(-INFINITY);
    for (long long i = tid; i < E; i += stride) {
        __builtin_prefetch((const void*)(p + (i + stride) * 2), 0, 0);
        v4f a = __builtin_nontemporal_load(p + i * 2);
        v4f b = __builtin_nontemporal_load(p + i * 2 + 1);
        m0 = __builtin_elementwise_max(m0, a);
        m1 = __builtin_elementwise_max(m1, b);
    }

    float mm[N_HEADS] = {m0.x, m0.y, m0.z, m0.w, m1.x, m1.y, m1.z, m1.w};
#pragma unroll
    for (int h = 0; h < N_HEADS; ++h) {
        float v = mm[h];
#pragma unroll
        for (int off = 16; off > 0; off >>= 1)   // wave32 butterfly reduction
            v = fmaxf(v, __shfl_xor(v, off, 32));
        if ((threadIdx.x & 31) == 0)
            atomicMax(&keys[h], enc_f32(v));     // native u32 atomic max
    }
}

// ----------------------------------------------------------- scale ----------
__global__ void k_scale(const unsigned* __restrict__ keys, float* __restrict__ scale) {
    int h = threadIdx.x;
    if (h < N_HEADS) {
        float m = dec_f32(keys[h]);
        float k = 0.0f;
        if (m > 10.0f) k = ceilf(log2f(fmaxf(m, 1e-30f) / 10.0f));
        k = fmaxf(k, 0.0f);
        scale[h] = exp2f(-k);
    }
}

// ---------------------------------------------- exp + segment-sum (atomic) --
__global__ void k_accum(const float* __restrict__ ev, const long long* __restrict__ row,
                        const float* __restrict__ scale, float* __restrict__ node_sum, int E) {
    const v4f s0 = *(const v4f*)(scale);
    const v4f s1 = *(const v4f*)(scale + 4);
    const long long tid = blockIdx.x * blockDim.x + threadIdx.x;
    const long long stride = (long long)gridDim.x * blockDim.x;
    const v4f* p = (const v4f*)ev;

    for (long long i = tid; i < E; i += stride) {
        __builtin_prefetch((const void*)(p + (i + stride) * 2), 0, 0);
        v4f a = __builtin_nontemporal_load(p + i * 2);
        v4f b = __builtin_nontemporal_load(p + i * 2 + 1);
        long long r = __builtin_nontemporal_load(row + i);
        a *= s0; b *= s1;
        float* dst = node_sum + r * N_HEADS;  // 3.2 MB accumulator: L2-resident
        unsafeAtomicAdd(dst + 0, expf(a.x));  // native global_atomic_add_f32
        unsafeAtomicAdd(dst + 1, expf(a.y));
        unsafeAtomicAdd(dst + 2, expf(a.z));
        unsafeAtomicAdd(dst + 3, expf(a.w));
        unsafeAtomicAdd(dst + 4, expf(b.x));
        unsafeAtomicAdd(dst + 5, expf(b.y));
        unsafeAtomicAdd(dst + 6, expf(b.z));
        unsafeAtomicAdd(dst + 7, expf(b.w));
    }
}

// -------------------------------------------------------- normalize ---------
__global__ void k_final(const float* __restrict__ ev, const long long* __restrict__ row,
                        const float* __restrict__ scale, const float* __restrict__ node_sum,
                        float* __restrict__ out, int E) {
    const v4f s0 = *(const v4f*)(scale);
    const v4f s1 = *(const v4f*)(scale + 4);
    const long long tid = blockIdx.x * blockDim.x + threadIdx.x;
    const long long stride = (long long)gridDim.x * blockDim.x;
    const v4f* p = (const v4f*)ev;
    v4f* q = (v4f*)out;

    for (long long i = tid; i < E; i += stride) {
        __builtin_prefetch((const void*)(p + (i + stride) * 2), 0, 0);
        v4f a = __builtin_nontemporal_load(p + i * 2);
        v4f b = __builtin_nontemporal_load(p + i * 2 + 1);
        long long r = __builtin_nontemporal_load(row + i);
        const v4f* ns = (const v4f*)(node_sum + r * N_HEADS);
        v4f d0 = ns[0];            // regular (temporal) load: hits L2
        v4f d1 = ns[1];
        a *= s0; b *= s1;
        v4f e0, e1;
        e0.x = expf(a.x) / d0.x;
        e0.y = expf(a.y) / d0.y;
        e0.z = expf(a.z) / d0.z;
        e0.w = expf(a.w) / d0.w;
        e1.x = expf(b.x) / d1.x;
        e1.y = expf(b.y) / d1.y;
        e1.z = expf(b.z) / d1.z;
        e1.w = expf(b.w) / d1.w;
        __builtin_nontemporal_store(e0, q + i * 2);
        __builtin_nontemporal_store(e1, q + i * 2 + 1);
    }
}

// ---------------------------------------------------------------------------
extern "C" void kernel_launch(void* const* d_in, const int* in_sizes, int n_in,
                              void* d_out, int out_size, void* d_ws, size_t ws_size,
                              hipStream_t stream) {
    const float* ev = (const float*)d_in[0];
    const long long* row = (const long long*)d_in[1];
    const int E = in_sizes[1];          // number of edges
    float* out = (float*)d_out;

    // Workspace layout (16B-aligned sections):
    //   [0, N*H*4)           node_sum  (3.2 MB)
    //   [+0,  +32)           head-max keys (8 x u32)
    //   [+32, +64)           scale     (8 x f32)
    char* ws = (char*)d_ws;
    float* node_sum = (float*)ws;
    unsigned* keys = (unsigned*)(ws + (size_t)N_NODES * N_HEADS * sizeof(float));
    float* scale = (float*)(keys + N_HEADS);

    const int initN = N_NODES * N_HEADS;
    k_init<<<(initN + BLOCK - 1) / BLOCK, BLOCK, 0, stream>>>(node_sum, keys, initN);
    k_max<<<STREAM_BLOCKS, BLOCK, 0, stream>>>(ev, keys, E);
    k_scale<<<1, 32, 0, stream>>>(keys, scale);
    k_accum<<<STREAM_BLOCKS, BLOCK, 0, stream>>>(ev, row, scale, node_sum, E);
    k_final<<<STREAM_BLOCKS, BLOCK, 0, stream>>>(ev, row, scale, node_sum, out, E);
}